// CLI_m_v1_63702954934483
// MI455X (gfx1250) — compile-verified
//
#include <hip/hip_runtime.h>

#define NA 12288
#define NB 12288
#define DTILES (NB / 16)   // 768 column tiles in the distance scan
#define XSTRIDE 260        // 256 + 4 pad -> conflict-free ds_load_b64 A-operand reads

typedef __attribute__((ext_vector_type(2))) float v2f;
typedef __attribute__((ext_vector_type(8))) float v8f;

// Branchless insert of (d, j) into ascending sorted 3-list.
__device__ __forceinline__ void top3_insert(float d, int j,
                                            float& s0, float& s1, float& s2,
                                            int& i0, int& i1, int& i2) {
  bool c2 = d < s2;
  float t2 = c2 ? d : s2;  int u2 = c2 ? j : i2;
  bool c1 = t2 < s1;
  float ns1 = c1 ? t2 : s1; float ns2 = c1 ? s1 : t2;
  int   nu1 = c1 ? u2 : i1; int   nu2 = c1 ? i1 : u2;
  bool c0 = ns1 < s0;
  float fs0 = c0 ? ns1 : s0; float fs1 = c0 ? s0 : ns1;
  int   fu0 = c0 ? nu1 : i0; int   fu1 = c0 ? i0 : nu1;
  s0 = fs0; s1 = fs1; s2 = ns2; i0 = fu0; i1 = fu1; i2 = nu2;
}

// ---------------------------------------------------------------------------
// Kernel 1: pack voxel coords for WMMA distance, precompute norms, transpose W
//   xa4[i] = (-2x,-2y,-2z, 1), na[i] = |x|^2
//   xb4[j] = ( x,  y,  z, |x|^2)
//   => wmma(A,B)[i][j] = -2*dot(xa,xb) + |xb|^2 ; + na[i] gives |xa-xb|^2
// ---------------------------------------------------------------------------
__global__ __launch_bounds__(256) void prep_kernel(
    const int* __restrict__ coords_a, const int* __restrict__ coords_b,
    const float* __restrict__ W1, const float* __restrict__ W2,
    float4* __restrict__ xa4, float* __restrict__ na,
    float4* __restrict__ xb4,
    float* __restrict__ W1T, float* __restrict__ W2T) {
  int id = blockIdx.x * 256 + threadIdx.x;   // grid covers [0, 65536)
  if (id < NA) {
    float x = (float)(coords_a[id * 3 + 0] >> 4);
    float y = (float)(coords_a[id * 3 + 1] >> 4);
    float z = (float)(coords_a[id * 3 + 2] >> 4);
    xa4[id] = make_float4(-2.f * x, -2.f * y, -2.f * z, 1.f);
    na[id] = x * x + y * y + z * z;
  }
  if (id < NB) {
    float x = (float)(coords_b[id * 3 + 0] >> 4);
    float y = (float)(coords_b[id * 3 + 1] >> 4);
    float z = (float)(coords_b[id * 3 + 2] >> 4);
    xb4[id] = make_float4(x, y, z, x * x + y * y + z * z);
  }
  {
    int n = id >> 8, k = id & 255;           // W[k][n] -> WT[n][k]
    W1T[id] = W1[k * 256 + n];
    W2T[id] = W2[k * 256 + n];
  }
}

// ---------------------------------------------------------------------------
// Kernel 2: brute-force masked top-3 per row via V_WMMA_F32_16X16X4_F32.
//   One block = 16 rows of A; 8 waves stripe the 768 column tiles.
//   Each lane keeps per-(row, column-residue) top-3 in registers; LDS merge.
// ---------------------------------------------------------------------------
__global__ __launch_bounds__(256) void topk_kernel(
    const float4* __restrict__ xa4, const float4* __restrict__ xb4,
    const float* __restrict__ na,
    const int* __restrict__ batch_a, const int* __restrict__ batch_b,
    float* __restrict__ kdw, int* __restrict__ kidx) {
  __shared__ float sd[16][8][16][3];  // 24 KB
  __shared__ int   si[16][8][16][3];  // 24 KB
  const int tid  = threadIdx.x;
  const int lane = tid & 31;
  const int w    = tid >> 5;          // wave id 0..7
  const int half = lane >> 4;         // 0: K0/K1 ; 1: K2/K3  (A/B operand split)
  const int l16  = lane & 15;
  const int rowbase = blockIdx.x * 16;

  // A operand (fixed for the whole scan): lane holds row rowbase+l16,
  // components (x,y) for lanes 0-15, (z,w) for lanes 16-31.
  const float* pa = (const float*)(xa4 + rowbase + l16) + half * 2;
  v2f a; a.x = pa[0]; a.y = pa[1];

  // Per-row constants for the 8 C-matrix VGPRs (rows v + 8*half).
  float nav[8]; int bav[8];
#pragma unroll
  for (int v = 0; v < 8; ++v) {
    int r = rowbase + v + 8 * half;
    nav[v] = na[r];
    bav[v] = batch_a[r];
  }

  float s0[8], s1[8], s2[8]; int i0[8], i1[8], i2[8];
#pragma unroll
  for (int v = 0; v < 8; ++v) {
    s0[v] = 1e30f; s1[v] = 1e30f; s2[v] = 1e30f;
    i0[v] = 0; i1[v] = 0; i2[v] = 0;
  }

  for (int t = w; t < DTILES; t += 8) {      // wave-uniform loop: EXEC all 1s
    int col = t * 16 + l16;
    const float* pb = (const float*)(xb4 + col) + half * 2;
    v2f b; b.x = pb[0]; b.y = pb[1];
    int bb = batch_b[col];
    v8f c = {0.f, 0.f, 0.f, 0.f, 0.f, 0.f, 0.f, 0.f};
    c = __builtin_amdgcn_wmma_f32_16x16x4_f32(false, a, false, b,
                                              (short)0, c, false, false);
#pragma unroll
    for (int v = 0; v < 8; ++v) {
      float d = c[v] + nav[v];               // |xa-xb|^2 (unscaled)
      d = (bb == bav[v]) ? d : 1e30f;        // exact cross-batch mask
      top3_insert(d, col, s0[v], s1[v], s2[v], i0[v], i1[v], i2[v]);
    }
  }

  // Dump per-lane candidates and merge 8*16*3 = 384 per row.
#pragma unroll
  for (int v = 0; v < 8; ++v) {
    int r = v + 8 * half;
    sd[r][w][l16][0] = s0[v]; si[r][w][l16][0] = i0[v];
    sd[r][w][l16][1] = s1[v]; si[r][w][l16][1] = i1[v];
    sd[r][w][l16][2] = s2[v]; si[r][w][l16][2] = i2[v];
  }
  __syncthreads();

  if (tid < 16) {
    int r = tid;
    float m0 = 1e30f, m1 = 1e30f, m2 = 1e30f;
    int j0 = 0, j1 = 0, j2 = 0;
    const float* pd = &sd[r][0][0][0];
    const int*   pi = &si[r][0][0][0];
    for (int q = 0; q < 384; ++q)
      top3_insert(pd[q], pi[q], m0, m1, m2, j0, j1, j2);
    float dm[3] = {m0, m1, m2};
    int   jm[3] = {j0, j1, j2};
#pragma unroll
    for (int k = 0; k < 3; ++k) {
      float dd = fmaxf(dm[k], 0.f) * (1.f / 16384.f);  // clip>=0, /128^2
      float dwv = 0.5f - fminf(dd, 0.5f);              // R - clip(d,0,R)
      kdw[(rowbase + r) * 3 + k]  = dwv;
      kidx[(rowbase + r) * 3 + k] = jm[k];
    }
  }
}

// ---------------------------------------------------------------------------
// Kernel 3: gather + f32 WMMA GEMM x2 + fuse + concat.
//   Block = 16 points -> 48 GEMM rows = 3 row-tiles. LDS buffer reused
//   X -> H -> G2. Wave w owns ntiles {2w,2w+1} x 3 row-tiles (6 accumulators).
// ---------------------------------------------------------------------------
__global__ __launch_bounds__(256) void fuse_gemm_kernel(
    const float* __restrict__ feats_a, const float* __restrict__ feats_b,
    const float* __restrict__ W1T, const float* __restrict__ bias1,
    const float* __restrict__ W2T, const float* __restrict__ bias2,
    const float* __restrict__ kdw, const int* __restrict__ kidx,
    float* __restrict__ out) {
  extern __shared__ float sm[];
  float* Xs   = sm;                       // 48 * XSTRIDE floats
  float* dwr  = sm + 48 * XSTRIDE;        // 48
  int*   jrow = (int*)(sm + 48 * XSTRIDE + 48);  // 48

  const int tid  = threadIdx.x;
  const int base = blockIdx.x * 16;       // first point index

  if (tid < 48) {
    dwr[tid]  = kdw[base * 3 + tid];
    jrow[tid] = kidx[base * 3 + tid];
  }
  __syncthreads();

  // Gather X[r][c] = feats_b[idx] * dw
  for (int e = tid; e < 48 * 256; e += 256) {
    int r = e >> 8, cc = e & 255;
    Xs[r * XSTRIDE + cc] = feats_b[jrow[r] * 256 + cc] * dwr[r];
  }
  __syncthreads();

  const int lane = tid & 31, w = tid >> 5;
  const int half = lane >> 4, l16 = lane & 15;
  const int n0 = 2 * w, n1 = 2 * w + 1;   // this wave's column tiles

  const float* pB10 = W1T + (n0 * 16 + l16) * 256 + half * 2;
  const float* pB11 = W1T + (n1 * 16 + l16) * 256 + half * 2;
  const float* pB20 = W2T + (n0 * 16 + l16) * 256 + half * 2;
  const float* pB21 = W2T + (n1 * 16 + l16) * 256 + half * 2;
  float bb10 = bias1[n0 * 16 + l16];
  float bb11 = bias1[n1 * 16 + l16];

  // ---- layer 1: H = relu(X @ W1 + b1) * dw ----
  v8f acc[3][2];
#pragma unroll
  for (int rt = 0; rt < 3; ++rt)
#pragma unroll
    for (int nt = 0; nt < 2; ++nt)
      acc[rt][nt] = (v8f){0.f, 0.f, 0.f, 0.f, 0.f, 0.f, 0.f, 0.f};

  for (int ks = 0; ks < 64; ++ks) {
    int k0 = ks * 4 + half * 2;
    v2f a0 = *(const v2f*)(Xs + (0 * 16 + l16) * XSTRIDE + k0);
    v2f a1 = *(const v2f*)(Xs + (1 * 16 + l16) * XSTRIDE + k0);
    v2f a2 = *(const v2f*)(Xs + (2 * 16 + l16) * XSTRIDE + k0);
    v2f b0 = *(const v2f*)(pB10 + ks * 4);
    v2f b1 = *(const v2f*)(pB11 + ks * 4);
    acc[0][0] = __builtin_amdgcn_wmma_f32_16x16x4_f32(false, a0, false, b0, (short)0, acc[0][0], false, false);
    acc[0][1] = __builtin_amdgcn_wmma_f32_16x16x4_f32(false, a0, false, b1, (short)0, acc[0][1], false, false);
    acc[1][0] = __builtin_amdgcn_wmma_f32_16x16x4_f32(false, a1, false, b0, (short)0, acc[1][0], false, false);
    acc[1][1] = __builtin_amdgcn_wmma_f32_16x16x4_f32(false, a1, false, b1, (short)0, acc[1][1], false, false);
    acc[2][0] = __builtin_amdgcn_wmma_f32_16x16x4_f32(false, a2, false, b0, (short)0, acc[2][0], false, false);
    acc[2][1] = __builtin_amdgcn_wmma_f32_16x16x4_f32(false, a2, false, b1, (short)0, acc[2][1], false, false);
  }
  __syncthreads();  // all X reads done before overwrite

#pragma unroll
  for (int rt = 0; rt < 3; ++rt)
#pragma unroll
    for (int nt = 0; nt < 2; ++nt) {
      int ncol = (nt ? n1 : n0) * 16 + l16;
      float bb = nt ? bb11 : bb10;
#pragma unroll
      for (int v = 0; v < 8; ++v) {
        int row = rt * 16 + v + 8 * half;
        float h = fmaxf(acc[rt][nt][v] + bb, 0.f) * dwr[row];
        Xs[row * XSTRIDE + ncol] = h;
      }
    }
  __syncthreads();

  // ---- layer 2: G2 = H @ W2 ----
#pragma unroll
  for (int rt = 0; rt < 3; ++rt)
#pragma unroll
    for (int nt = 0; nt < 2; ++nt)
      acc[rt][nt] = (v8f){0.f, 0.f, 0.f, 0.f, 0.f, 0.f, 0.f, 0.f};

  for (int ks = 0; ks < 64; ++ks) {
    int k0 = ks * 4 + half * 2;
    v2f a0 = *(const v2f*)(Xs + (0 * 16 + l16) * XSTRIDE + k0);
    v2f a1 = *(const v2f*)(Xs + (1 * 16 + l16) * XSTRIDE + k0);
    v2f a2 = *(const v2f*)(Xs + (2 * 16 + l16) * XSTRIDE + k0);
    v2f b0 = *(const v2f*)(pB20 + ks * 4);
    v2f b1 = *(const v2f*)(pB21 + ks * 4);
    acc[0][0] = __builtin_amdgcn_wmma_f32_16x16x4_f32(false, a0, false, b0, (short)0, acc[0][0], false, false);
    acc[0][1] = __builtin_amdgcn_wmma_f32_16x16x4_f32(false, a0, false, b1, (short)0, acc[0][1], false, false);
    acc[1][0] = __builtin_amdgcn_wmma_f32_16x16x4_f32(false, a1, false, b0, (short)0, acc[1][0], false, false);
    acc[1][1] = __builtin_amdgcn_wmma_f32_16x16x4_f32(false, a1, false, b1, (short)0, acc[1][1], false, false);
    acc[2][0] = __builtin_amdgcn_wmma_f32_16x16x4_f32(false, a2, false, b0, (short)0, acc[2][0], false, false);
    acc[2][1] = __builtin_amdgcn_wmma_f32_16x16x4_f32(false, a2, false, b1, (short)0, acc[2][1], false, false);
  }
  __syncthreads();  // all H reads done

#pragma unroll
  for (int rt = 0; rt < 3; ++rt)
#pragma unroll
    for (int nt = 0; nt < 2; ++nt) {
      int ncol = (nt ? n1 : n0) * 16 + l16;
#pragma unroll
      for (int v = 0; v < 8; ++v) {
        int row = rt * 16 + v + 8 * half;
        Xs[row * XSTRIDE + ncol] = acc[rt][nt][v];
      }
    }
  __syncthreads();

  // ---- fuse: sum over k (3 rows) + 3*b2 (b2 broadcasts before the k-sum),
  //      concat with feats_a ----
  for (int e = tid; e < 16 * 256; e += 256) {
    int il = e >> 8, cc = e & 255;
    int i = base + il;
    float f = Xs[(il * 3 + 0) * XSTRIDE + cc] +
              Xs[(il * 3 + 1) * XSTRIDE + cc] +
              Xs[(il * 3 + 2) * XSTRIDE + cc] + 3.f * bias2[cc];
    out[i * 512 + cc]       = feats_a[i * 256 + cc];
    out[i * 512 + 256 + cc] = f;
  }
}

// ---------------------------------------------------------------------------
extern "C" void kernel_launch(void* const* d_in, const int* in_sizes, int n_in,
                              void* d_out, int out_size, void* d_ws, size_t ws_size,
                              hipStream_t stream) {
  const int*   batch_a  = (const int*)  d_in[0];
  const int*   coords_a = (const int*)  d_in[1];
  const float* feats_a  = (const float*)d_in[2];
  const int*   batch_b  = (const int*)  d_in[3];
  const int*   coords_b = (const int*)  d_in[4];
  const float* feats_b  = (const float*)d_in[5];
  const float* W1 = (const float*)d_in[6];
  const float* b1 = (const float*)d_in[7];
  const float* W2 = (const float*)d_in[8];
  const float* b2 = (const float*)d_in[9];
  float* out = (float*)d_out;

  // Workspace layout (floats): ~1.26 MB total
  float*  ws  = (float*)d_ws;
  float4* xa4 = (float4*)(ws);                 // NA*4
  float4* xb4 = (float4*)(ws + NA * 4);        // NB*4
  float*  na  = ws + NA * 4 + NB * 4;          // NA
  float*  W1T = na + NA;                       // 65536
  float*  W2T = W1T + 65536;                   // 65536
  float*  kdw = W2T + 65536;                   // NA*3
  int*    kidx = (int*)(kdw + NA * 3);         // NA*3

  prep_kernel<<<256, 256, 0, stream>>>(coords_a, coords_b, W1, W2,
                                       xa4, na, xb4, W1T, W2T);
  topk_kernel<<<NA / 16, 256, 0, stream>>>(xa4, xb4, na, batch_a, batch_b,
                                           kdw, kidx);
  size_t smem = (size_t)(48 * XSTRIDE + 96) * sizeof(float);
  fuse_gemm_kernel<<<NA / 16, 256, smem, stream>>>(feats_a, feats_b,
                                                   W1T, b1, W2T, b2,
                                                   kdw, kidx, out);
}